// SelfCorrelationPercPooling_77979426226648
// MI455X (gfx1250) — compile-verified
//
#include <hip/hip_runtime.h>

// CDNA5 / gfx1250. wave32. Fused self-correlation + descending sort + percentile gather.
// x: [4, 64, 64, 256] fp32  ->  out: [4, 64, 64, 256] fp32
//
// Grid: 1024 workgroups (4 batches x 256 row-strips of 16 rows).
// Each workgroup:
//   Phase 1: corr strip [16 x 4096] = X[m0:m0+16,:] * X^T / 256 via V_WMMA_F32_16X16X4_F32,
//            accumulated fp32, written to LDS (256 KB).
//   Phase 2: cooperative bitonic sort (descending) of all 16 rows in LDS.
//   Phase 3: gather 256 percentile ranks round(linspace(1,4095,256)) -> global.

typedef float v2f __attribute__((ext_vector_type(2)));
typedef float v8f __attribute__((ext_vector_type(8)));

#define M_TOT 4096   // H*W
#define F_DIM 256    // feature dim (= K of the GEMM)
#define ROWS  16     // rows of corr per workgroup (one WMMA M-tile)
#define NPOOL 256

__global__ __launch_bounds__(256) void
self_corr_perc_pool_kernel(const float* __restrict__ x, float* __restrict__ out) {
    extern __shared__ float s_corr[];            // [ROWS][M_TOT] = 256 KB

    const int tid  = threadIdx.x;
    const int wave = tid >> 5;                   // 8 waves
    const int lane = tid & 31;

    const int b  = blockIdx.x >> 8;              // batch
    const int m0 = (blockIdx.x & 255) << 4;      // first of 16 rows

    const float* Xb = x + (size_t)b * M_TOT * F_DIM;

    // ---- Phase 1: WMMA fp32 GEMM strip -> LDS ----
    // A-frag (16x4 f32): lane<16 -> row=lane, K={0,1}; lane>=16 -> row=lane-16, K={2,3}.
    // B-frag (4x16) for X*X^T has the identical per-lane addressing, based at n0.
    const int half = lane >> 4;                  // 0 or 1
    const int lrow = lane & 15;
    const float* arow = Xb + (size_t)(m0 + lrow) * F_DIM + 2 * half;

    for (int nt = wave; nt < (M_TOT / 16); nt += 8) {
        const int n0 = nt << 4;
        const float* brow = Xb + (size_t)(n0 + lrow) * F_DIM + 2 * half;

        v8f c = {};
#pragma unroll 8
        for (int kk = 0; kk < F_DIM; kk += 4) {
            v2f a  = *(const v2f*)(arow + kk);   // 8B-aligned global_load_b64
            v2f bb = *(const v2f*)(brow + kk);
            // (neg_a, A, neg_b, B, c_mod, C, reuse_a, reuse_b)
            c = __builtin_amdgcn_wmma_f32_16x16x4_f32(
                    false, a, false, bb, (short)0, c, false, false);
        }

        // C/D layout: lane<16 -> (M=j, N=lane); lane>=16 -> (M=8+j, N=lane-16)
        const int crow0 = half << 3;
#pragma unroll
        for (int j = 0; j < 8; ++j)
            s_corr[(crow0 + j) * M_TOT + n0 + lrow] = c[j] * (1.0f / (float)F_DIM);
    }
    __syncthreads();

    // ---- Phase 2: descending bitonic sort, 16 rows in parallel ----
    // 16 rows * 2048 pairs = 32768 compare-exchanges per stage; 128 per thread.
    for (int k = 2; k <= M_TOT; k <<= 1) {
        for (int j = k >> 1; j > 0; j >>= 1) {
            for (int p = tid; p < ROWS * (M_TOT / 2); p += 256) {
                const int row = p >> 11;                    // /2048
                const int idx = p & (M_TOT / 2 - 1);
                const int i   = ((idx & ~(j - 1)) << 1) | (idx & (j - 1));
                const int ip  = i | j;
                float* rp = s_corr + row * M_TOT;
                const float va = rp[i];
                const float vb = rp[ip];
                const bool keepMaxAtI = ((i & k) == 0);     // descending overall
                if ((va < vb) == keepMaxAtI) { rp[i] = vb; rp[ip] = va; }
            }
            __syncthreads();
        }
    }

    // ---- Phase 3: percentile gather ----
    // ranks = round(linspace(1, 4095, 256)); rintf = round-nearest-even (matches jnp.round)
    const int rank = (int)rintf(1.0f + (float)tid * (4094.0f / 255.0f));
#pragma unroll
    for (int r = 0; r < ROWS; ++r) {
        const size_t o = (((size_t)b * M_TOT + (size_t)(m0 + r)) << 8) + (size_t)tid;
        out[o] = s_corr[r * M_TOT + rank];
    }
}

extern "C" void kernel_launch(void* const* d_in, const int* in_sizes, int n_in,
                              void* d_out, int out_size, void* d_ws, size_t ws_size,
                              hipStream_t stream) {
    (void)in_sizes; (void)n_in; (void)d_ws; (void)ws_size; (void)out_size;
    const float* x = (const float*)d_in[0];
    float* out = (float*)d_out;

    const int n_blocks = 4 * (M_TOT / ROWS);     // 1024
    const size_t lds_bytes = (size_t)ROWS * M_TOT * sizeof(float);  // 262144 B (<= 320KB WGP LDS)
    self_corr_perc_pool_kernel<<<dim3(n_blocks), dim3(256), lds_bytes, stream>>>(x, out);
}